// CRF_15272903705189
// MI455X (gfx1250) — compile-verified
//
#include <hip/hip_runtime.h>
#include <stdint.h>

// Problem constants
#define BB     128
#define LL     4096
#define TT     48
#define LP     4094              // L-2 (pot time length)
#define NSTEP  4093              // forward transition steps
#define CHUNK  256               // traceback chunk (steps) per LDS buffer
#define NCHUNK ((NSTEP + CHUNK - 1) / CHUNK)   // 16

// d_out float-element layout: [pot | transitions | tags]
#define POT_N    ((size_t)BB * LP * TT)        // 25,153,536 floats
#define TR_OFF   POT_N
#define TAGS_OFF (TR_OFF + (size_t)TT * TT)

// Scratch aliased into the pot region (pot is written LAST, after traceback):
//   bytes [0, BB*NSTEP*TT)          : backpointers, u8, [b][t][j]
//   bytes [BB*NSTEP*TT, +BB*4)      : last_tag int per batch
#define BP_BYTES      ((size_t)BB * NSTEP * TT)   // 25,147,392 (<< 100.6 MB pot)
#define LT_OFF_BYTES  BP_BYTES

// ---------------- CDNA5 async global->LDS helpers ----------------
// Param 1 of the builtin (per hipcc diagnostic) is a global (AS1, printed
// '__device__') pointer to a GCC-vector int4 (the 16-byte b128 payload).
typedef int v4i __attribute__((vector_size(16)));
typedef __attribute__((address_space(1))) v4i* gv4i_p;   // global
typedef __attribute__((address_space(3))) v4i* lv4i_p;   // LDS

__device__ __forceinline__ void async_load_b128(void* lds, const void* gsrc) {
#if __has_builtin(__builtin_amdgcn_global_load_async_to_lds_b128)
  __builtin_amdgcn_global_load_async_to_lds_b128(
      (gv4i_p)(uint64_t)(uintptr_t)gsrc,    // global source (AS1), const dropped
      (lv4i_p)(uint32_t)(uintptr_t)lds,     // LDS dest: low 32 bits == LDS offset
      0, 0);                                // imm offset, cpol
#else
  uint32_t l = (uint32_t)(uintptr_t)lds;
  uint64_t g = (uint64_t)(uintptr_t)gsrc;
  asm volatile("global_load_async_to_lds_b128 %0, %1, off"
               :: "v"(l), "v"(g) : "memory");
#endif
}

__device__ __forceinline__ void wait_async0() {
#if __has_builtin(__builtin_amdgcn_s_wait_asynccnt)
  __builtin_amdgcn_s_wait_asynccnt(0);
#else
  asm volatile("s_wait_asynccnt 0x0" ::: "memory");
#endif
}

__device__ __forceinline__ void wait_async1() {
#if __has_builtin(__builtin_amdgcn_s_wait_asynccnt)
  __builtin_amdgcn_s_wait_asynccnt(1);
#else
  asm volatile("s_wait_asynccnt 0x1" ::: "memory");
#endif
}

// ---------------- Forward Viterbi: one workgroup (2 waves) per batch ----------------
__global__ __launch_bounds__(64) void crf_forward(const float* __restrict__ x,
                                                  const float* __restrict__ tr,
                                                  unsigned char* __restrict__ bp_out,
                                                  int* __restrict__ last_tag) {
  __shared__ __align__(16) float s_state[TT];
  __shared__ __align__(16) float s_emit[2][TT];   // double-buffered emissions

  const int b = blockIdx.x;
  const int j = threadIdx.x;                      // output tag index (j < 48 active)
  const float* xb = x + (size_t)b * (LL * TT);

  // Transition column T[:, j] lives in 48 VGPRs for the whole scan.
  float tc[TT];
  if (j < TT) {
#pragma unroll
    for (int i = 0; i < TT; ++i) tc[i] = tr[i * TT + j];
    s_state[j]   = xb[TT + j];        // first = pot[:,0,:] = inputs[:,1,:]
    s_emit[0][j] = xb[2 * TT + j];    // emit for step t=0 = inputs[:,2,:]
  }
  __syncthreads();

  unsigned char* bpb = bp_out + (size_t)b * NSTEP * TT;

  for (int t = 0; t < NSTEP; ++t) {
    // Kick async DMA of next step's 192-byte emission row into the other buffer.
    // (t+3 <= 4095 always, stays in bounds; issuers are in wave 0 only.)
    if (threadIdx.x < 12) {
      async_load_b128(&s_emit[(t + 1) & 1][threadIdx.x * 4],
                      xb + (size_t)(t + 3) * TT + (size_t)threadIdx.x * 4);
    }

    float m = 0.0f;
    int   bp = 0;
    if (j < TT) {
      // Broadcast-read state via 12 x ds_load_b128.
      float st[TT];
#pragma unroll
      for (int q = 0; q < TT / 4; ++q) {
        float4 v = ((const float4*)s_state)[q];
        st[4*q+0] = v.x; st[4*q+1] = v.y; st[4*q+2] = v.z; st[4*q+3] = v.w;
      }
      m = st[0] + tc[0];
#pragma unroll
      for (int i = 1; i < TT; ++i) {
        float s = st[i] + tc[i];
        if (s > m) { m = s; bp = i; }   // strict '>' => first-max, matches jnp.argmax
      }
      bpb[(size_t)t * TT + j] = (unsigned char)bp;   // fire-and-forget u8 store
    }

    // Older async (this step's emission, issued last iter) must be retired;
    // the one just issued may stay in flight => wait ASYNCcnt <= 1, in-order.
    wait_async1();
    __syncthreads();                    // publish DMA'd emissions to both waves

    if (j < TT) s_state[j] = m + s_emit[t & 1][j];
    __syncthreads();
  }

  if (threadIdx.x == 0) {
    float m = s_state[0]; int a = 0;
#pragma unroll
    for (int i = 1; i < TT; ++i) if (s_state[i] > m) { m = s_state[i]; a = i; }
    last_tag[b] = a;
  }
}

// ---------------- Traceback: chunked LDS staging, double-buffered async DMA ----------------
__global__ __launch_bounds__(64) void crf_backward(const unsigned char* __restrict__ bp,
                                                   const int* __restrict__ last_tag,
                                                   int* __restrict__ out_tags) {
  __shared__ __align__(16) unsigned char s_bp[2][CHUNK * TT];  // 2 x 12 KB

  const int b = blockIdx.x;
  const unsigned char* bpb = bp + (size_t)b * NSTEP * TT;
  int* tb = out_tags + (size_t)b * LP;

  // Preload the top (partial) chunk.
  {
    const int k = NCHUNK - 1;
    const int t0 = k * CHUNK;
    const int n = (NSTEP - t0) * 3;                 // rows*48/16 b128 transfers
    for (int q = threadIdx.x; q < n; q += 64)
      async_load_b128(&s_bp[k & 1][q * 16], bpb + (size_t)t0 * TT + (size_t)q * 16);
  }
  wait_async0();
  __syncthreads();

  int tag = 0;
  if (threadIdx.x == 0) {
    tag = last_tag[b];
    tb[LP - 1] = tag;                               // tags[:, 4093] = last_tag
  }

  for (int k = NCHUNK - 1; k >= 0; --k) {
    const int t0 = k * CHUNK;
    const int rows = min(NSTEP - t0, CHUNK);

    // Prefetch the next-lower chunk while thread 0 traces the current one.
    if (k > 0) {
      const int t0p = (k - 1) * CHUNK;
      for (int q = threadIdx.x; q < CHUNK * 3; q += 64)
        async_load_b128(&s_bp[(k - 1) & 1][q * 16],
                        bpb + (size_t)t0p * TT + (size_t)q * 16);
    }

    if (threadIdx.x == 0) {
      const unsigned char* buf = s_bp[k & 1];
      for (int lt = rows - 1; lt >= 0; --lt) {      // dependent ds_load_u8 chain
        int prev = buf[lt * TT + tag];
        tb[t0 + lt] = prev;
        tag = prev;
      }
    }

    wait_async0();                                  // next chunk landed
    __syncthreads();
  }
}

// ---------------- Output copies (run LAST: they overwrite the bp scratch) ----------------
__global__ __launch_bounds__(256) void copy_pot(const float4* __restrict__ src,
                                                float4* __restrict__ dst) {
  const int b = blockIdx.y;
  const int i = blockIdx.x * 256 + threadIdx.x;
  const int PB = LP * TT / 4;                       // 49,128 float4 per batch
  if (i < PB)
    dst[(size_t)b * PB + i] = src[(size_t)b * (LL * TT / 4) + (TT / 4) + i];
}

__global__ __launch_bounds__(256) void copy_tr(const float* __restrict__ tr,
                                               float* __restrict__ dst) {
  const int i = blockIdx.x * 256 + threadIdx.x;
  if (i < TT * TT) dst[i] = tr[i];
}

// ---------------- Host launcher ----------------
extern "C" void kernel_launch(void* const* d_in, const int* in_sizes, int n_in,
                              void* d_out, int out_size, void* d_ws, size_t ws_size,
                              hipStream_t stream) {
  (void)in_sizes; (void)n_in; (void)out_size; (void)d_ws; (void)ws_size;
  const float* x  = (const float*)d_in[0];   // inputs (128,4096,48) f32
  const float* tr = (const float*)d_in[1];   // transition_params (48,48) f32
  float* out = (float*)d_out;

  unsigned char* bp_scratch = (unsigned char*)d_out;                // aliases pot region
  int* last_tag = (int*)((char*)d_out + LT_OFF_BYTES);
  int* tags     = (int*)(out + TAGS_OFF);                           // int32 bit pattern

  crf_forward <<<BB, 64, 0, stream>>>(x, tr, bp_scratch, last_tag);
  crf_backward<<<BB, 64, 0, stream>>>(bp_scratch, last_tag, tags);

  dim3 gp((LP * TT / 4 + 255) / 256, BB);                           // 192 x 128 blocks
  copy_pot<<<gp, 256, 0, stream>>>((const float4*)x, (float4*)out); // overwrites scratch
  copy_tr <<<(TT * TT + 255) / 256, 256, 0, stream>>>(tr, out + TR_OFF);
}